// HypergraphConvolution_87445534147336
// MI455X (gfx1250) — compile-verified
//
#include <hip/hip_runtime.h>

// Hypergraph convolution: out = Dv^-1/2 H De^-1 H^T Dv^-1/2 X W + b
// N=32768, E=8192, F=64. H is 1 GiB f32 -> bandwidth floor = 3 passes over H.
// Matmuls: V_WMMA_F32_16X16X4_F32. H/Y tiles streamed into LDS with
// GLOBAL_LOAD_ASYNC_TO_LDS_B128 (ASYNCcnt) and double-buffered.

constexpr int   N_   = 32768;
constexpr int   E_   = 8192;
constexpr int   F_   = 64;
constexpr float EPS_ = 1e-5f;
constexpr int   LSTR = 68;   // LDS row stride in words; 272B = 16B-aligned rows

typedef float v2f __attribute__((ext_vector_type(2)));
typedef float v8f __attribute__((ext_vector_type(8)));

__device__ inline v8f vzero8() {
  v8f z;
  #pragma unroll
  for (int i = 0; i < 8; ++i) z[i] = 0.0f;
  return z;
}

// D = A(16x4) * B(4x16) + C, f32. Layouts per CDNA5 ISA 7.12.2:
//  A: lane L holds A[L%16][(L/16)*2 + {0,1}]
//  B: lane L holds B[(L/16)*2 + {0,1}][L%16]
//  C/D: VGPR r, lane L -> M = r + (L/16)*8, N = L%16
__device__ inline v8f wmma4(v2f a, v2f b, v8f c) {
  return __builtin_amdgcn_wmma_f32_16x16x4_f32(
      false, a, false, b, (short)0, c, false, false);
}

// --- CDNA5 async global->LDS (tracked by ASYNCcnt) ------------------------
// the vdst VGPR gives the LDS byte address; low 32 bits of a generic LDS pointer are
// the wave-relative LDS byte offset (ISA 10.2 aperture rules).
__device__ __forceinline__ void async_b128_nt(unsigned ldsOff, const float* g) {
  asm volatile("global_load_async_to_lds_b128 %0, %1, off th:TH_LOAD_NT"
               :: "v"(ldsOff), "v"(g) : "memory");
}
__device__ __forceinline__ void async_b128(unsigned ldsOff, const float* g) {
  asm volatile("global_load_async_to_lds_b128 %0, %1, off"
               :: "v"(ldsOff), "v"(g) : "memory");
}
#define WAIT_ASYNC(n) asm volatile("s_wait_asynccnt " #n ::: "memory")

__device__ __forceinline__ unsigned ldsOffOf(const float* p) {
  return (unsigned)(unsigned long long)p;
}

// ---------------------------------------------------------------------------
// Pass 1: fused row sums (deg_v) and column sums (deg_e) of H in ONE pass.
// Tile: 256 rows x 32 cols per block. grid = (E/32, N/256).
// ---------------------------------------------------------------------------
__global__ __launch_bounds__(256) void k_degrees(const float* __restrict__ H,
                                                 float* __restrict__ deg_v,
                                                 float* __restrict__ deg_e) {
  const int t = threadIdx.x;
  const int w = t >> 5, lane = t & 31;
  const int c0 = blockIdx.x * 32;
  const int r0 = blockIdx.y * 256;

  float colacc = 0.0f;
  for (int i = 0; i < 32; ++i) {
    const int r = r0 + i * 8 + w;
    const float v = H[(size_t)r * E_ + c0 + lane];
    colacc += v;
    float s = v;                       // wave32 tree reduce for the row sum
    s += __shfl_down(s, 16);
    s += __shfl_down(s, 8);
    s += __shfl_down(s, 4);
    s += __shfl_down(s, 2);
    s += __shfl_down(s, 1);
    if (lane == 0) unsafeAtomicAdd(&deg_v[r], s);
  }
  __shared__ float cpart[256];
  cpart[t] = colacc;
  __syncthreads();
  if (t < 32) {
    float s = 0.0f;
    #pragma unroll
    for (int j = 0; j < 8; ++j) s += cpart[j * 32 + t];
    unsafeAtomicAdd(&deg_e[c0 + t], s);
  }
}

// ---------------------------------------------------------------------------
// Pass 2 (small): Y1[n,f] = (1/sqrt(deg_v[n]+eps)) * sum_k X[n,k] W[k,f]
// One wave -> 16 rows x 64 cols. W staged transposed [f][k] in LDS.
// ---------------------------------------------------------------------------
__global__ __launch_bounds__(256) void k_xw(const float* __restrict__ X,
                                            const float* __restrict__ W,
                                            const float* __restrict__ deg_v,
                                            float* __restrict__ Y1) {
  __shared__ float lW[64 * LSTR];      // [f][k]
  const int t  = threadIdx.x;
  const int lc = t & 63, lr4 = t >> 6;
  #pragma unroll
  for (int i = 0; i < 16; ++i) {
    const int k = i * 4 + lr4;
    lW[lc * LSTR + k] = W[k * 64 + lc];
  }
  __syncthreads();

  const int w = t >> 5, lane = t & 31;
  const int m = lane & 15, kb = (lane >> 4) << 1;
  const int nw = blockIdx.x * 128 + w * 16;

  v8f acc[4];
  #pragma unroll
  for (int q = 0; q < 4; ++q) acc[q] = vzero8();

  for (int k = 0; k < 64; k += 4) {
    const v2f a = *(const v2f*)(X + (size_t)(nw + m) * 64 + k + kb);
    #pragma unroll
    for (int q = 0; q < 4; ++q) {
      const v2f b = *(const v2f*)(&lW[(q * 16 + m) * LSTR + k + kb]);
      acc[q] = wmma4(a, b, acc[q]);
    }
  }

  const int hi = lane >> 4, nn = lane & 15;
  #pragma unroll
  for (int r = 0; r < 8; ++r) {
    const int n = nw + r + hi * 8;
    const float s = 1.0f / sqrtf(deg_v[n] + EPS_);
    #pragma unroll
    for (int q = 0; q < 4; ++q)
      Y1[(size_t)n * 64 + q * 16 + nn] = acc[q][r] * s;
  }
}

// ---------------------------------------------------------------------------
// Pass 3 (big): Y2[e,f] += sum_n H[n,e] * Y1[n,f]   (H^T @ Y1)
// Block: 64e x 64f tile, K-split 8-way over N. Tiles kept in NATURAL row-major
// layout [n][e] / [n][f] so they can be filled by async b128 loads; fragments
// read as paired b32 one LDS row apart (-> ds_load_2addr_b32).
// Double-buffered: issue chunk c+1 asyncs, s_wait_asynccnt, barrier, compute.
// ---------------------------------------------------------------------------
__global__ __launch_bounds__(256) void k_gemm2(const float* __restrict__ H,
                                               const float* __restrict__ Y1,
                                               float* __restrict__ Y2) {
  __shared__ float lA[2][64 * LSTR];   // [k=n_local][e_local]
  __shared__ float lB[2][64 * LSTR];   // [k=n_local][f]
  const int t = threadIdx.x;
  const int w = t >> 5, lane = t & 31;
  const int m = lane & 15, kb = (lane >> 4) << 1;
  const int esub = (w & 3) * 16, fh = (w >> 2) * 32;
  const int e0 = blockIdx.x * 64;
  const int nBeg = blockIdx.y * (N_ / 8);

  // async loader mapping: 4 rows-groups of 16, 16B column blocks
  const int c4 = t & 15;               // 16B column block
  const int rB = t >> 4;               // base row 0..15
  const unsigned ldsRow = (unsigned)(rB * LSTR * 4 + c4 * 16);

  v8f acc0 = vzero8(), acc1 = vzero8();
  constexpr int C = (N_ / 8) / 64;     // 64 chunks

  {
    const int n = nBeg + rB;
    const unsigned a0 = ldsOffOf(&lA[0][0]) + ldsRow;
    const unsigned b0 = ldsOffOf(&lB[0][0]) + ldsRow;
    #pragma unroll
    for (int i = 0; i < 4; ++i) {
      async_b128_nt(a0 + i * 16u * LSTR * 4u, H  + (size_t)(n + i * 16) * E_ + e0 + c4 * 4);
      async_b128  (b0 + i * 16u * LSTR * 4u, Y1 + (size_t)(n + i * 16) * 64 + c4 * 4);
    }
  }

  for (int c = 0; c < C; ++c) {
    const int s = c & 1;
    if (c + 1 < C) {
      const int n = nBeg + (c + 1) * 64 + rB;
      const unsigned a0 = ldsOffOf(&lA[s ^ 1][0]) + ldsRow;
      const unsigned b0 = ldsOffOf(&lB[s ^ 1][0]) + ldsRow;
      #pragma unroll
      for (int i = 0; i < 4; ++i) {
        async_b128_nt(a0 + i * 16u * LSTR * 4u, H  + (size_t)(n + i * 16) * E_ + e0 + c4 * 4);
        async_b128  (b0 + i * 16u * LSTR * 4u, Y1 + (size_t)(n + i * 16) * 64 + c4 * 4);
      }
      WAIT_ASYNC(8);
    } else {
      WAIT_ASYNC(0);
    }
    __syncthreads();
    #pragma unroll
    for (int k = 0; k < 64; k += 4) {
      v2f a, b0, b1;
      a.x  = lA[s][(k + kb) * LSTR + esub + m];
      a.y  = lA[s][(k + kb + 1) * LSTR + esub + m];
      b0.x = lB[s][(k + kb) * LSTR + fh + m];
      b0.y = lB[s][(k + kb + 1) * LSTR + fh + m];
      b1.x = lB[s][(k + kb) * LSTR + fh + 16 + m];
      b1.y = lB[s][(k + kb + 1) * LSTR + fh + 16 + m];
      acc0 = wmma4(a, b0, acc0);
      acc1 = wmma4(a, b1, acc1);
    }
    __syncthreads();
  }

  const int hi = lane >> 4, nn = lane & 15;
  #pragma unroll
  for (int r = 0; r < 8; ++r) {
    const int e = e0 + esub + r + hi * 8;
    unsafeAtomicAdd(&Y2[(size_t)e * 64 + fh + nn],      acc0[r]);
    unsafeAtomicAdd(&Y2[(size_t)e * 64 + fh + 16 + nn], acc1[r]);
  }
}

// ---------------------------------------------------------------------------
// Pass 4 (tiny): Y3 = De^-1 * Y2
// ---------------------------------------------------------------------------
__global__ __launch_bounds__(256) void k_scale(const float* __restrict__ Y2,
                                               const float* __restrict__ deg_e,
                                               float* __restrict__ Y3) {
  const int idx = blockIdx.x * 256 + threadIdx.x;
  Y3[idx] = Y2[idx] / (deg_e[idx >> 6] + EPS_);
}

// ---------------------------------------------------------------------------
// Pass 5 (big): Y4[n,f] += sum_e H[n,e] * Y3[e,f]   (H @ Y3)
// A tile natural [n][e] (M=n, K=e -> contiguous K pairs, b64 frag reads),
// B tile natural [e][f] (paired-b32 frags). K-split 4-way over E.
// ---------------------------------------------------------------------------
__global__ __launch_bounds__(256) void k_gemm3(const float* __restrict__ H,
                                               const float* __restrict__ Y3,
                                               float* __restrict__ Y4) {
  __shared__ float lA[2][64 * LSTR];   // [n_local][k=e_local]
  __shared__ float lB[2][64 * LSTR];   // [k=e_local][f]
  const int t = threadIdx.x;
  const int w = t >> 5, lane = t & 31;
  const int m = lane & 15, kb = (lane >> 4) << 1;
  const int nsub = (w & 3) * 16, fh = (w >> 2) * 32;
  const int n0 = blockIdx.x * 64;
  const int eBeg = blockIdx.y * (E_ / 4);

  const int c4 = t & 15;
  const int rB = t >> 4;
  const unsigned ldsRow = (unsigned)(rB * LSTR * 4 + c4 * 16);

  v8f acc0 = vzero8(), acc1 = vzero8();
  constexpr int C = (E_ / 4) / 64;     // 32 chunks

  {
    const unsigned a0 = ldsOffOf(&lA[0][0]) + ldsRow;
    const unsigned b0 = ldsOffOf(&lB[0][0]) + ldsRow;
    #pragma unroll
    for (int i = 0; i < 4; ++i) {
      async_b128_nt(a0 + i * 16u * LSTR * 4u, H  + (size_t)(n0 + rB + i * 16) * E_ + eBeg + c4 * 4);
      async_b128  (b0 + i * 16u * LSTR * 4u, Y3 + (size_t)(eBeg + rB + i * 16) * 64 + c4 * 4);
    }
  }

  for (int c = 0; c < C; ++c) {
    const int s = c & 1;
    if (c + 1 < C) {
      const int ec = eBeg + (c + 1) * 64;
      const unsigned a0 = ldsOffOf(&lA[s ^ 1][0]) + ldsRow;
      const unsigned b0 = ldsOffOf(&lB[s ^ 1][0]) + ldsRow;
      #pragma unroll
      for (int i = 0; i < 4; ++i) {
        async_b128_nt(a0 + i * 16u * LSTR * 4u, H  + (size_t)(n0 + rB + i * 16) * E_ + ec + c4 * 4);
        async_b128  (b0 + i * 16u * LSTR * 4u, Y3 + (size_t)(ec + rB + i * 16) * 64 + c4 * 4);
      }
      WAIT_ASYNC(8);
    } else {
      WAIT_ASYNC(0);
    }
    __syncthreads();
    #pragma unroll
    for (int k = 0; k < 64; k += 4) {
      const v2f a = *(const v2f*)(&lA[s][(nsub + m) * LSTR + k + kb]);
      v2f b0, b1;
      b0.x = lB[s][(k + kb) * LSTR + fh + m];
      b0.y = lB[s][(k + kb + 1) * LSTR + fh + m];
      b1.x = lB[s][(k + kb) * LSTR + fh + 16 + m];
      b1.y = lB[s][(k + kb + 1) * LSTR + fh + 16 + m];
      acc0 = wmma4(a, b0, acc0);
      acc1 = wmma4(a, b1, acc1);
    }
    __syncthreads();
  }

  const int hi = lane >> 4, nn = lane & 15;
  #pragma unroll
  for (int r = 0; r < 8; ++r) {
    const int n = n0 + nsub + r + hi * 8;
    unsafeAtomicAdd(&Y4[(size_t)n * 64 + fh + nn],      acc0[r]);
    unsafeAtomicAdd(&Y4[(size_t)n * 64 + fh + 16 + nn], acc1[r]);
  }
}

// ---------------------------------------------------------------------------
// Pass 6: out = Dv^-1/2 * Y4 + b
// ---------------------------------------------------------------------------
__global__ __launch_bounds__(256) void k_final(const float* __restrict__ Y4,
                                               const float* __restrict__ deg_v,
                                               const float* __restrict__ b,
                                               float* __restrict__ out) {
  const int idx = blockIdx.x * 256 + threadIdx.x;
  const int n = idx >> 6, f = idx & 63;
  out[idx] = Y4[idx] * (1.0f / sqrtf(deg_v[n] + EPS_)) + b[f];
}

extern "C" void kernel_launch(void* const* d_in, const int* in_sizes, int n_in,
                              void* d_out, int out_size, void* d_ws, size_t ws_size,
                              hipStream_t stream) {
  const float* X = (const float*)d_in[0];   // [N, 64]
  const float* H = (const float*)d_in[1];   // [N, E]
  const float* W = (const float*)d_in[2];   // [64, 64]
  const float* b = (const float*)d_in[3];   // [64]
  float* out = (float*)d_out;               // [N, 64]

  float* ws    = (float*)d_ws;              // ~21 MB total
  float* deg_v = ws;                         // N
  float* deg_e = deg_v + N_;                 // E
  float* Y1    = deg_e + E_;                 // N*64
  float* Y2    = Y1 + (size_t)N_ * 64;       // E*64
  float* Y3    = Y2 + (size_t)E_ * 64;       // E*64
  float* Y4    = Y3 + (size_t)E_ * 64;       // N*64

  hipMemsetAsync(deg_v, 0, (size_t)(N_ + E_) * sizeof(float), stream);
  hipMemsetAsync(Y2,    0, (size_t)E_ * 64 * sizeof(float), stream);
  hipMemsetAsync(Y4,    0, (size_t)N_ * 64 * sizeof(float), stream);

  k_degrees<<<dim3(E_ / 32, N_ / 256), 256, 0, stream>>>(H, deg_v, deg_e);
  k_xw     <<<dim3(N_ / 128),          256, 0, stream>>>(X, W, deg_v, Y1);
  k_gemm2  <<<dim3(E_ / 64, 8),        256, 0, stream>>>(H, Y1, Y2);
  k_scale  <<<dim3(E_ * 64 / 256),     256, 0, stream>>>(Y2, deg_e, Y3);
  k_gemm3  <<<dim3(N_ / 64, 4),        256, 0, stream>>>(H, Y3, Y4);
  k_final  <<<dim3((N_ * 64) / 256),   256, 0, stream>>>(Y4, deg_v, b, out);
}